// FFA_65051574665856
// MI455X (gfx1250) — compile-verified
//
#include <hip/hip_runtime.h>
#include <math.h>

typedef __attribute__((ext_vector_type(2))) float v2f;
typedef __attribute__((ext_vector_type(8))) float v8f;

#define NB 8
#define NC 512
#define NH 224
#define NW 224
#define NHEADS 8
#define HD 64
#define WIN 7
#define HWP 1024        // 32*32 pooled tokens
#define EPS 1e-5f

// workspace layout (float offsets)
#define Y_OFF    0ull                         // NB*NC*HWP = 4,194,304
#define YSUM_OFF (Y_OFF + 4194304ull)         // NB*NC     = 4096
#define SQ_OFF   (YSUM_OFF + 4096ull)         // NB*NC     = 4096
#define STAT_OFF (SQ_OFF + 4096ull)           // 2*NB      = 16
#define S_OFF    (STAT_OFF + 16ull)           // NB*NHEADS*HD*HD = 262,144
#define GATE_OFF (S_OFF + 262144ull)          // NB*NC     = 4096

// ---------------------------------------------------------------------------
// Kernel 1: 7x7 avg-pool, one block per (b,c) plane; also per-(b,c) sum/sumsq
// ---------------------------------------------------------------------------
__global__ void ffa_pool_kernel(const float* __restrict__ x, float* __restrict__ ws) {
    const int blk = blockIdx.x;            // b*NC + c
    const int tid = threadIdx.x;           // 256 threads
    const float* xp = x + (size_t)blk * (NH * NW);
    float* yp = ws + Y_OFF + (size_t)blk * HWP;

    float s = 0.f, sq = 0.f;
    #pragma unroll
    for (int i = 0; i < 4; ++i) {
        const int p  = tid + i * 256;      // 0..1023
        const int oh = p >> 5;
        const int ow = p & 31;
        const float* base = xp + oh * WIN * NW + ow * WIN;
        float acc = 0.f;
        #pragma unroll
        for (int r = 0; r < WIN; ++r) {
            const float* row = base + r * NW;
            #pragma unroll
            for (int cc = 0; cc < WIN; ++cc) acc += row[cc];
        }
        acc *= (1.0f / 49.0f);
        yp[p] = acc;
        s  += acc;
        sq += acc * acc;
    }

    __shared__ float red_s[256];
    __shared__ float red_q[256];
    red_s[tid] = s; red_q[tid] = sq;
    __syncthreads();
    for (int off = 128; off > 0; off >>= 1) {
        if (tid < off) { red_s[tid] += red_s[tid + off]; red_q[tid] += red_q[tid + off]; }
        __syncthreads();
    }
    if (tid == 0) {
        ws[YSUM_OFF + blk] = red_s[0];
        ws[SQ_OFF   + blk] = red_q[0];
    }
}

// ---------------------------------------------------------------------------
// Kernel 2: per-batch mean / rsqrt(var+eps), deterministic reduction
// ---------------------------------------------------------------------------
__global__ void ffa_stats_kernel(float* __restrict__ ws) {
    const int b   = blockIdx.x;            // 8 blocks
    const int tid = threadIdx.x;           // 256 threads
    float s = 0.f, sq = 0.f;
    for (int c = tid; c < NC; c += 256) {
        s  += ws[YSUM_OFF + b * NC + c];
        sq += ws[SQ_OFF   + b * NC + c];
    }
    __shared__ float red_s[256];
    __shared__ float red_q[256];
    red_s[tid] = s; red_q[tid] = sq;
    __syncthreads();
    for (int off = 128; off > 0; off >>= 1) {
        if (tid < off) { red_s[tid] += red_s[tid + off]; red_q[tid] += red_q[tid + off]; }
        __syncthreads();
    }
    if (tid == 0) {
        const float n   = (float)(NC * HWP);
        const float mu  = red_s[0] / n;
        const float var = red_q[0] / n - mu * mu;
        ws[STAT_OFF + 2 * b]     = mu;
        ws[STAT_OFF + 2 * b + 1] = rsqrtf(var + EPS);
    }
}

// ---------------------------------------------------------------------------
// Kernel 3: Gram matrix S[d,e] = sum_n z[d,n]*z[e,n] per (b,head) via
//           V_WMMA_F32_16X16X4_F32. 16 waves/block, one 16x16 tile per wave.
// ---------------------------------------------------------------------------
#define CHUNK 128
#define LDP   (CHUNK + 4)   // pad: stride 132 % 64 banks = 4 -> conflict-free

__global__ void ffa_gram_kernel(const float* __restrict__ gamma,
                                const float* __restrict__ beta,
                                float* __restrict__ ws) {
    const int bh = blockIdx.x;             // b*NHEADS + h, 64 blocks
    const int b  = bh / NHEADS;
    const int h  = bh % NHEADS;
    const int tid = threadIdx.x;           // 512 threads = 16 waves

    __shared__ float zs[HD * LDP];
    __shared__ float ascale[HD];
    __shared__ float bshift[HD];

    const float mu = ws[STAT_OFF + 2 * b];
    const float rs = ws[STAT_OFF + 2 * b + 1];
    if (tid < HD) {
        const int c = h * HD + tid;
        const float a = rs * gamma[c];
        ascale[tid] = a;
        bshift[tid] = beta[c] - mu * a;    // z = y*a + bshift
    }

    const int wave  = tid >> 5;
    const int lane  = tid & 31;
    const int dtile = wave >> 2;           // row tile 0..3
    const int etile = wave & 3;            // col tile 0..3
    const int half  = lane >> 4;           // K-pair select (ISA 16x4 layout)
    const int l16   = lane & 15;
    const int mrow  = dtile * 16 + l16;    // A-matrix row (Z row)
    const int erow  = etile * 16 + l16;    // B-matrix col (Z row, transposed product)

    const float* yb = ws + Y_OFF + (size_t)(b * NC + h * HD) * HWP;

    v8f acc = {};
    for (int ch = 0; ch < HWP / CHUNK; ++ch) {
        __syncthreads();
        // stage 64ch x 128n chunk of z into LDS: 8192 floats / 512 thr = 16 ea
        #pragma unroll
        for (int i = 0; i < 16; ++i) {
            const int idx = tid + i * 512;
            const int cc  = idx >> 7;
            const int nn  = idx & 127;
            zs[cc * LDP + nn] =
                yb[(size_t)cc * HWP + ch * CHUNK + nn] * ascale[cc] + bshift[cc];
        }
        __syncthreads();
        #pragma unroll
        for (int it = 0; it < CHUNK / 4; ++it) {
            const int kb = it * 4 + half * 2;
            v2f av, bv;
            av[0] = zs[mrow * LDP + kb];
            av[1] = zs[mrow * LDP + kb + 1];
            bv[0] = zs[erow * LDP + kb];
            bv[1] = zs[erow * LDP + kb + 1];
            acc = __builtin_amdgcn_wmma_f32_16x16x4_f32(
                false, av, false, bv, (short)0, acc, false, false);
        }
    }

    // C/D layout: VGPR i -> M = i + half*8, N = l16 (per 16x16 f32 table)
    float* sp = ws + S_OFF + (size_t)bh * (HD * HD);
    #pragma unroll
    for (int i = 0; i < 8; ++i) {
        const int m = dtile * 16 + half * 8 + i;
        const int n = etile * 16 + l16;
        sp[m * HD + n] = acc[i];
    }
}

// ---------------------------------------------------------------------------
// Kernel 4: softmax over S rows, dot with vmean, sigmoid -> gate[b,c]
// ---------------------------------------------------------------------------
__global__ void ffa_attn_gate_kernel(const float* __restrict__ gamma,
                                     const float* __restrict__ beta,
                                     const float* __restrict__ wq,
                                     const float* __restrict__ wk,
                                     const float* __restrict__ wv,
                                     float* __restrict__ ws) {
    const int bh = blockIdx.x;             // 64 blocks
    const int b  = bh / NHEADS;
    const int h  = bh % NHEADS;
    const int d  = threadIdx.x;            // 64 threads

    __shared__ float vm[HD];
    __shared__ float wkl[HD];

    const float mu = ws[STAT_OFF + 2 * b];
    const float rs = ws[STAT_OFF + 2 * b + 1];
    const int   c  = h * HD + d;

    // mean_n yn[c] = (mean_n y[c] - mu)*rs*gamma + beta ; vmean = wv * that
    const float ymean = ws[YSUM_OFF + b * NC + c] * (1.0f / HWP);
    const float ynm   = (ymean - mu) * rs * gamma[c] + beta[c];
    vm[d]  = wv[c] * ynm;
    wkl[d] = wk[c];
    __syncthreads();

    const float* srow = ws + S_OFF + (size_t)bh * (HD * HD) + d * HD;
    const float wqd = wq[c] * 0.125f;      // scale = head_dim^-0.5 = 1/8

    float l[HD];
    float mx = -1e30f;
    #pragma unroll
    for (int e = 0; e < HD; ++e) {
        l[e] = wqd * wkl[e] * srow[e];
        mx = fmaxf(mx, l[e]);
    }
    float sum = 0.f;
    #pragma unroll
    for (int e = 0; e < HD; ++e) { l[e] = __expf(l[e] - mx); sum += l[e]; }
    float o = 0.f;
    #pragma unroll
    for (int e = 0; e < HD; ++e) o += l[e] * vm[e];
    o /= sum;
    ws[GATE_OFF + b * NC + c] = 1.0f / (1.0f + __expf(-o));
}

// ---------------------------------------------------------------------------
// Kernel 5: out = gate[b,c] * x  (pure HBM bandwidth, float4)
// grid.y = b*NC plane index (no integer division), grid.x*256 = 12544 float4s
// ---------------------------------------------------------------------------
__global__ void ffa_apply_kernel(const float* __restrict__ x,
                                 const float* __restrict__ ws,
                                 float* __restrict__ out) {
    const int bc = blockIdx.y;
    const int i  = blockIdx.x * 256 + threadIdx.x;   // 49*256 = 12544 exactly
    const float g = ws[GATE_OFF + bc];
    const size_t base = (size_t)bc * (NH * NW);      // 50176, 16B-aligned
    const float4* x4 = (const float4*)(x + base);
    float4*       o4 = (float4*)(out + base);
    float4 v = x4[i];
    v.x *= g; v.y *= g; v.z *= g; v.w *= g;
    o4[i] = v;
}

// ---------------------------------------------------------------------------
extern "C" void kernel_launch(void* const* d_in, const int* in_sizes, int n_in,
                              void* d_out, int out_size, void* d_ws, size_t ws_size,
                              hipStream_t stream) {
    (void)in_sizes; (void)n_in; (void)out_size; (void)ws_size;
    const float* x     = (const float*)d_in[0];
    const float* gamma = (const float*)d_in[1];
    const float* beta  = (const float*)d_in[2];
    const float* wq    = (const float*)d_in[3];
    const float* wk    = (const float*)d_in[4];
    const float* wv    = (const float*)d_in[5];
    float* out = (float*)d_out;
    float* ws  = (float*)d_ws;

    ffa_pool_kernel     <<<NB * NC,           256, 0, stream>>>(x, ws);
    ffa_stats_kernel    <<<NB,                256, 0, stream>>>(ws);
    ffa_gram_kernel     <<<NB * NHEADS,       512, 0, stream>>>(gamma, beta, ws);
    ffa_attn_gate_kernel<<<NB * NHEADS,       HD,  0, stream>>>(gamma, beta, wq, wk, wv, ws);
    ffa_apply_kernel    <<<dim3(49, NB * NC), 256, 0, stream>>>(x, ws, out);
}